// GINConv_layer_25031069401546
// MI455X (gfx1250) — compile-verified
//
#include <hip/hip_runtime.h>
#include <math.h>

// ---------------- problem constants ----------------
#define N_NODES 100000
#define N_EDGES 1600000
#define DIM     128
#define ND      (N_NODES * DIM)     // 12.8M floats per activation buffer
#define LN_EPS  1e-5f

typedef __attribute__((ext_vector_type(2))) float v2f;
typedef __attribute__((ext_vector_type(8))) float v8f;

// Fire-and-forget FP32 atomic add (L2 atomic unit). Avoids the CAS loop the
// compiler may emit for atomicAdd(float*). S_ENDPGM implies wait-idle, so all
// atomics are complete at kernel boundary (stream ordering gives us the sync).
__device__ __forceinline__ void atomic_fadd(float* p, float v) {
    asm volatile("global_atomic_add_f32 %0, %1, off" :: "v"(p), "v"(v) : "memory");
}

// Async global->LDS copy, 16B per lane, tracked by ASYNCcnt (CDNA5 path).
// VDST operand = VGPR holding the LDS byte address (low 32 bits of the flat
// shared-pointer), VADDR = 64-bit global address.
__device__ __forceinline__ void async_copy_b128(uint32_t lds_addr, const void* gaddr) {
    asm volatile("global_load_async_to_lds_b128 %0, %1, off"
                 :: "v"(lds_addr), "v"(gaddr) : "memory");
}
__device__ __forceinline__ void wait_asynccnt0() {
    asm volatile("s_wait_asynccnt 0x0" ::: "memory");
}

// ---------------- kernel 1: zero accumulator + stats ----------------
__global__ void k_zero(float* __restrict__ buf, float* __restrict__ stats) {
    int i = blockIdx.x * blockDim.x + threadIdx.x;      // 12500*256 threads, float4 each
    float4 z = {0.f, 0.f, 0.f, 0.f};
    ((float4*)buf)[i] = z;
    if (blockIdx.x == 0 && threadIdx.x < 16) stats[threadIdx.x] = 0.f;
}

// ---------------- kernel 2: edge scatter-add (GIN aggregation) --------------
// One wave32 per edge; lane handles 4 consecutive features (float4 gather,
// 4 fp32 atomics). Both node table (51MB) and aggr (51MB) are L2-resident
// (192MB L2), so this phase runs at L2 rather than HBM speed after warmup.
__global__ void k_scatter(const float* __restrict__ node,
                          const int*   __restrict__ src,
                          const int*   __restrict__ dst,
                          float*       __restrict__ aggr) {
    int wave = blockIdx.x * 8 + (threadIdx.x >> 5);
    int lane = threadIdx.x & 31;
    int s = src[wave];
    int t = dst[wave];
    float4 v = ((const float4*)(node + (size_t)s * DIM))[lane];
    float* p = aggr + (size_t)t * DIM + lane * 4;
    atomic_fadd(p + 0, v.x);
    atomic_fadd(p + 1, v.y);
    atomic_fadd(p + 2, v.z);
    atomic_fadd(p + 3, v.w);
}

// ---------------- kernel 3: fused (pre-op -> GEMM 128x128 -> bias -> stats) -
// MODE 0: A = (1+eps)*node + aggr          (GIN combine)
// MODE 1/2: A = relu((X - mean)*rstd*ln_w[c] + ln_b[c])   (graph-LN + ReLU)
// Block = 256 thr = 8 waves; block computes 16 rows x 128 cols of Y.
// Wave w owns N-tile [16w,16w+16); K=128 swept with 32x V_WMMA_F32_16X16X4_F32.
//
// Data movement: W (64KB) is DMA'd into LDS with GLOBAL_LOAD_ASYNC_TO_LDS_B128
// *overlapped* with the A-tile transform (global loads + VALU); the WMMA loop
// then touches only LDS. A tile padded to stride 132 (132%64banks=4) so the
// 16-lane column reads are bank-conflict-free.
template <int MODE>
__global__ void k_gemm(const float* __restrict__ X,        // MODE 1/2 input
                       const float* __restrict__ node,     // MODE 0
                       const float* __restrict__ aggr,     // MODE 0
                       const float* __restrict__ epsp,     // MODE 0 (1 elem)
                       const float* __restrict__ lnw,      // MODE 1/2
                       const float* __restrict__ lnb,      // MODE 1/2
                       const float* __restrict__ stats_in, // [0]=mean [1]=rstd
                       const float* __restrict__ W,        // 128x128 row-major
                       const float* __restrict__ bias,     // 128
                       float*       __restrict__ Y,
                       float*       __restrict__ stats_out) // [0]+=sum [1]+=sumsq
{
    __shared__ __align__(16) float Ws[DIM * DIM];  // 64KB staged weights
    __shared__ __align__(16) float As[16 * 132];   // 16x128 A tile, padded
    __shared__ float red[512];                     // block reduction scratch

    const int tid  = threadIdx.x;
    const int row0 = blockIdx.x * 16;

    // ---- kick off async DMA of W into LDS (16 float4 per thread) ----
    {
        uint32_t lds_base = (uint32_t)(uintptr_t)Ws;
        const float4* Wg = (const float4*)W;       // 4096 float4s
#pragma unroll
        for (int j = 0; j < 16; ++j) {
            int idx = tid + 256 * j;
            async_copy_b128(lds_base + (uint32_t)idx * 16u, Wg + idx);
        }
    }

    float mean = 0.f, rstd = 0.f, onepe = 0.f;
    if (MODE == 0) {
        onepe = 1.0f + epsp[0];
    } else {
        mean = stats_in[0];
        rstd = stats_in[1];
    }

    // ---- A-tile transform (overlaps with the async W DMA) ----
    for (int i = tid; i < 16 * 128; i += 256) {
        int r = i >> 7, c = i & 127;
        size_t g = (size_t)(row0 + r) * DIM + c;
        float v;
        if (MODE == 0) {
            v = onepe * node[g] + aggr[g];
        } else {
            v = (X[g] - mean) * rstd * lnw[c] + lnb[c];
            v = v > 0.f ? v : 0.f;                       // ReLU
        }
        As[r * 132 + c] = v;
    }

    wait_asynccnt0();        // this wave's DMA chunk is in LDS
    __syncthreads();         // everyone's chunk visible

    const int wv   = tid >> 5;
    const int lane = tid & 31;
    const int half = lane >> 4;      // which 16-lane half
    const int l16  = lane & 15;
    const int n0   = wv * 16;        // this wave's N-tile

    // 16x16 f32 accumulator = 8 VGPRs (ISA 7.12.2 layout).
    v8f acc = {0.f, 0.f, 0.f, 0.f, 0.f, 0.f, 0.f, 0.f};

    for (int kb = 0; kb < DIM; kb += 4) {
        int k = kb + 2 * half;
        // A frag (16x4): VGPR0 = K(k), VGPR1 = K(k+1); one b64 LDS read
        v2f a = *(const v2f*)&As[l16 * 132 + k];
        // B frag (4x16): Ws[k][n0+l16], Ws[k+1][n0+l16] from LDS
        v2f b;
        b.x = Ws[k * DIM + n0 + l16];
        b.y = Ws[(k + 1) * DIM + n0 + l16];
        acc = __builtin_amdgcn_wmma_f32_16x16x4_f32(
                  false, a, false, b, (short)0, acc, false, false);
    }

    // Epilogue: + bias, store, and accumulate LN statistics.
    float bb = bias[n0 + l16];
    float ls = 0.f, lq = 0.f;
#pragma unroll
    for (int r = 0; r < 8; ++r) {
        float v = acc[r] + bb;
        size_t row = (size_t)(row0 + r + 8 * half);
        Y[row * DIM + n0 + l16] = v;
        ls += v;
        lq += v * v;
    }

    red[tid] = ls;
    red[256 + tid] = lq;
    __syncthreads();
    for (int s = 128; s > 0; s >>= 1) {
        if (tid < s) {
            red[tid]       += red[tid + s];
            red[256 + tid] += red[256 + tid + s];
        }
        __syncthreads();
    }
    if (tid == 0) {
        atomic_fadd(&stats_out[0], red[0]);
        atomic_fadd(&stats_out[1], red[256]);
    }
}

// ---------------- kernel 4: stats finalize (1 thread) ----------------
// graph LN: x' = x - mean; std = sqrt(mean(x'^2)) = sqrt(E[x^2]-mean^2)
__global__ void k_finalize(float* __restrict__ stats) {
    const float invnd = 1.0f / (float)ND;
    float mean = stats[0] * invnd;
    float var  = stats[1] * invnd - mean * mean;
    var = var > 0.f ? var : 0.f;
    stats[2] = mean;
    stats[3] = 1.0f / (sqrtf(var) + LN_EPS);
}

// ---------------- kernel 5: final LN + ReLU ----------------
__global__ void k_final(const float* __restrict__ X,
                        const float* __restrict__ stats,  // [2]=mean [3]=rstd
                        const float* __restrict__ w,
                        const float* __restrict__ b,
                        float* __restrict__ out) {
    int i = blockIdx.x * blockDim.x + threadIdx.x;   // float4 per thread
    float mean = stats[2], rstd = stats[3];
    float4 x = ((const float4*)X)[i];
    int c = (i * 4) & 127;                            // 4 consecutive cols
    float4 o;
    o.x = fmaxf((x.x - mean) * rstd * w[c + 0] + b[c + 0], 0.f);
    o.y = fmaxf((x.y - mean) * rstd * w[c + 1] + b[c + 1], 0.f);
    o.z = fmaxf((x.z - mean) * rstd * w[c + 2] + b[c + 2], 0.f);
    o.w = fmaxf((x.w - mean) * rstd * w[c + 3] + b[c + 3], 0.f);
    ((float4*)out)[i] = o;
}

// ---------------- launcher ----------------
extern "C" void kernel_launch(void* const* d_in, const int* in_sizes, int n_in,
                              void* d_out, int out_size, void* d_ws, size_t ws_size,
                              hipStream_t stream) {
    const float* node = (const float*)d_in[0];
    const int*   eidx = (const int*)  d_in[1];
    const float* epsp = (const float*)d_in[4];
    const float* W1   = (const float*)d_in[5];
    const float* b1   = (const float*)d_in[6];
    const float* ln1w = (const float*)d_in[7];
    const float* ln1b = (const float*)d_in[8];
    const float* W2   = (const float*)d_in[9];
    const float* b2   = (const float*)d_in[10];
    const float* ln2w = (const float*)d_in[11];
    const float* ln2b = (const float*)d_in[12];
    const float* W3   = (const float*)d_in[13];
    const float* b3   = (const float*)d_in[14];
    const float* lnow = (const float*)d_in[15];
    const float* lnob = (const float*)d_in[16];

    const int* src = eidx;              // edge_index[0]
    const int* dst = eidx + N_EDGES;    // edge_index[1]

    float* ws    = (float*)d_ws;
    float* bufA  = ws;                  // ND floats (aggr/h0, then t2)
    float* bufB  = ws + (size_t)ND;     // ND floats (t1, then t3)
    float* stats = ws + 2 * (size_t)ND; // 16 floats: 3 stages x {sum,sq,mean,rstd}
    float* out   = (float*)d_out;

    const int GEMM_BLOCKS = N_NODES / 16;  // 6250

    // 1) zero aggr + stats
    k_zero<<<ND / 1024, 256, 0, stream>>>(bufA, stats);
    // 2) scatter-add aggregation (1 wave per edge)
    k_scatter<<<N_EDGES / 8, 256, 0, stream>>>(node, src, dst, bufA);
    // 3) h @ W1 + b1 (A = (1+eps)*node + aggr), stats stage 0
    k_gemm<0><<<GEMM_BLOCKS, 256, 0, stream>>>(
        nullptr, node, bufA, epsp, nullptr, nullptr, nullptr,
        W1, b1, bufB, stats + 0);
    k_finalize<<<1, 1, 0, stream>>>(stats + 0);
    // 4) relu(LN(t1)) @ W2 + b2, stats stage 1
    k_gemm<1><<<GEMM_BLOCKS, 256, 0, stream>>>(
        bufB, nullptr, nullptr, nullptr, ln1w, ln1b, stats + 2,
        W2, b2, bufA, stats + 4);
    k_finalize<<<1, 1, 0, stream>>>(stats + 4);
    // 5) relu(LN(t2)) @ W3 + b3, stats stage 2
    k_gemm<2><<<GEMM_BLOCKS, 256, 0, stream>>>(
        bufA, nullptr, nullptr, nullptr, ln2w, ln2b, stats + 6,
        W3, b3, bufB, stats + 8);
    k_finalize<<<1, 1, 0, stream>>>(stats + 8);
    // 6) out = relu(LN(t3))
    k_final<<<ND / 1024, 256, 0, stream>>>(bufB, stats + 8, lnow, lnob, out);
}